// RGCNLayer_3066606649850
// MI455X (gfx1250) — compile-verified
//
#include <hip/hip_runtime.h>

#define B_ 4
#define S_ 256
#define H_ 768

typedef float v2f __attribute__((ext_vector_type(2)));
typedef float v8f __attribute__((ext_vector_type(8)));

// ---------- Stage 1: row/col relation histograms -> inverse neighbor counts ----------
__global__ void count_kernel(const int* __restrict__ rels,
                             float* __restrict__ invf,
                             float* __restrict__ invr) {
  __shared__ int cf[32];
  __shared__ int cr[32];
  int b = blockIdx.x / S_;
  int i = blockIdx.x % S_;
  int t = threadIdx.x;
  if (t < 32) { cf[t] = 0; cr[t] = 0; }
  __syncthreads();
  int rf = rels[(b * S_ + i) * S_ + t];   // row i, forward edges
  int rr = rels[(b * S_ + t) * S_ + i];   // column i, reverse edges
  atomicAdd(&cf[rf], 1);
  atomicAdd(&cr[rr], 1);
  __syncthreads();
  if (t < 64) {
    int r = t & 31;
    int cnt = (t < 32) ? cf[r] : cr[r];
    float inv = 1.0f / fmaxf((float)cnt, 1.0f);
    if (t < 32) invf[(b * S_ + i) * 32 + r] = inv;
    else        invr[(b * S_ + i) * 32 + r] = inv;
  }
}

// ---------- Stage 2: dense basis-coefficient adjacency W[b][c][i][j] ----------
__global__ void wmat_kernel(const int* __restrict__ rels,
                            const float* __restrict__ rel_weight,  // [64][4]
                            const float* __restrict__ invf,
                            const float* __restrict__ invr,
                            float* __restrict__ Wm) {               // [B][4][S][S]
  int idx = blockIdx.x * blockDim.x + threadIdx.x;  // < B*S*S
  int b = idx >> 16;
  int rem = idx & 65535;
  int i = rem >> 8;
  int j = rem & 255;
  int rf = rels[(b * S_ + i) * S_ + j];
  int rr = rels[(b * S_ + j) * S_ + i];
  float sf = (rf > 0) ? invf[(b * S_ + i) * 32 + rf] : 0.0f;
  float sr = (rr > 0) ? invr[(b * S_ + i) * 32 + rr] : 0.0f;
#pragma unroll
  for (int c = 0; c < 4; ++c) {
    float w = 0.0f;
    if (rf > 0) w += rel_weight[rf * 4 + c] * sf;
    if (rr > 0) w += rel_weight[(rr + 32) * 4 + c] * sr;
    Wm[(((b * 4 + c) * S_) + i) * S_ + j] = w;
  }
}

// ---------- self_weight (out,in) -> transposed (in,out) ----------
__global__ void transpose_kernel(const float* __restrict__ src,
                                 float* __restrict__ dst) {
  __shared__ float tile[32][33];
  int x  = blockIdx.x * 32 + threadIdx.x;
  int y0 = blockIdx.y * 32 + threadIdx.y;
#pragma unroll
  for (int k = 0; k < 32; k += 8)
    tile[threadIdx.y + k][threadIdx.x] = src[(y0 + k) * H_ + x];
  __syncthreads();
  int xo  = blockIdx.y * 32 + threadIdx.x;
  int yo0 = blockIdx.x * 32 + threadIdx.y;
#pragma unroll
  for (int k = 0; k < 32; k += 8)
    dst[(yo0 + k) * H_ + xo] = tile[threadIdx.x][threadIdx.y + k];
}

// ---------- Generic 64x64-tile fp32 WMMA GEMM core ----------
// 128 threads = 4 waves; each wave -> 32x32 = 2x2 V_WMMA_F32_16X16X4_F32 tiles.
__device__ __forceinline__ void gemm64x64(const float* __restrict__ A, int lda,
                                          const float* __restrict__ B, int ldb,
                                          float* __restrict__ C, int ldc,
                                          int K, bool accumulate) {
  __shared__ float Alds[64][20];   // 64x16 tile, padded (stride 20 -> conflict-free frag reads)
  __shared__ float Blds[16][68];   // 16x64 tile, padded (row stride 272B, 16B aligned)

  const int t    = threadIdx.x;
  const int m0   = blockIdx.y * 64;
  const int n0   = blockIdx.x * 64;
  const int lane = t & 31;
  const int wave = t >> 5;
  const int lh   = lane & 15;   // lane within half
  const int sel  = lane >> 4;   // 0: lanes 0-15, 1: lanes 16-31
  const int wm   = (wave >> 1) * 32;
  const int wn   = (wave & 1) * 32;

  v8f acc[2][2];
#pragma unroll
  for (int ti = 0; ti < 2; ++ti)
#pragma unroll
    for (int tj = 0; tj < 2; ++tj)
      acc[ti][tj] = {};

  for (int kb = 0; kb < K; kb += 16) {
    __syncthreads();
    // A tile: 64x16 = 256 float4, 2 per thread
#pragma unroll
    for (int r = 0; r < 2; ++r) {
      int idx = t + r * 128;
      int row = idx >> 2;
      int c4  = (idx & 3) << 2;
      const float4 v = *reinterpret_cast<const float4*>(&A[(m0 + row) * (long)lda + kb + c4]);
      *reinterpret_cast<float4*>(&Alds[row][c4]) = v;
    }
    // B tile: 16x64 = 256 float4, 2 per thread
#pragma unroll
    for (int r = 0; r < 2; ++r) {
      int idx = t + r * 128;
      int row = idx >> 4;
      int c4  = (idx & 15) << 2;
      const float4 v = *reinterpret_cast<const float4*>(&B[(kb + row) * (long)ldb + n0 + c4]);
      *reinterpret_cast<float4*>(&Blds[row][c4]) = v;
    }
    __syncthreads();

#pragma unroll
    for (int kq = 0; kq < 4; ++kq) {
      const int k = kq * 4;
      v2f af[2], bf[2];
      // A fragment 16x4: lane = M, VGPR = K (high half-lanes get K+2)
#pragma unroll
      for (int ti = 0; ti < 2; ++ti) {
        int row = wm + ti * 16 + lh;
        af[ti].x = Alds[row][k + 2 * sel];
        af[ti].y = Alds[row][k + 2 * sel + 1];
      }
      // B fragment 4x16: lane = N, VGPR = K (same K split)
#pragma unroll
      for (int tj = 0; tj < 2; ++tj) {
        int col = wn + tj * 16 + lh;
        bf[tj].x = Blds[k + 2 * sel][col];
        bf[tj].y = Blds[k + 2 * sel + 1][col];
      }
#pragma unroll
      for (int ti = 0; ti < 2; ++ti)
#pragma unroll
        for (int tj = 0; tj < 2; ++tj)
          acc[ti][tj] = __builtin_amdgcn_wmma_f32_16x16x4_f32(
              false, af[ti], false, bf[tj], (short)0, acc[ti][tj], false, false);
    }
  }

  // Epilogue: C/D layout — VGPR v = row offset, high half-lanes M+8, lane = N
#pragma unroll
  for (int ti = 0; ti < 2; ++ti) {
#pragma unroll
    for (int tj = 0; tj < 2; ++tj) {
      int gr = m0 + wm + ti * 16 + sel * 8;
      int gc = n0 + wn + tj * 16 + lh;
#pragma unroll
      for (int v = 0; v < 8; ++v) {
        float* p = &C[(gr + v) * (long)ldc + gc];
        float val = acc[ti][tj][v];
        if (accumulate) *p = *p + val; else *p = val;
      }
    }
  }
}

// Z[b,c] = W[b,c] (256x256) @ H_b (256x768), stored as Zcat[b] row-major [256, 3072]
__global__ void gemm_stage1(const float* __restrict__ Wm,
                            const float* __restrict__ hidden,
                            float* __restrict__ Z) {
  int z = blockIdx.z;            // b*4 + c
  int b = z >> 2, c = z & 3;
  gemm64x64(Wm + (long)z * S_ * S_, S_,
            hidden + (long)b * S_ * H_, H_,
            Z + (long)b * S_ * 4 * H_ + c * H_, 4 * H_,
            S_, false);
}

// out_b = Zcat[b] (256x3072) @ basic_flat (3072x768)
__global__ void gemm_stage2(const float* __restrict__ Z,
                            const float* __restrict__ basic,
                            float* __restrict__ out) {
  int b = blockIdx.z;
  gemm64x64(Z + (long)b * S_ * 4 * H_, 4 * H_,
            basic, H_,
            out + (long)b * S_ * H_, H_,
            4 * H_, false);
}

// out_b += H_b (256x768) @ self_weight^T (768x768)
__global__ void gemm_stage3(const float* __restrict__ hidden,
                            const float* __restrict__ Wt,
                            float* __restrict__ out) {
  int b = blockIdx.z;
  gemm64x64(hidden + (long)b * S_ * H_, H_,
            Wt, H_,
            out + (long)b * S_ * H_, H_,
            H_, true);
}

extern "C" void kernel_launch(void* const* d_in, const int* in_sizes, int n_in,
                              void* d_out, int out_size, void* d_ws, size_t ws_size,
                              hipStream_t stream) {
  (void)in_sizes; (void)n_in; (void)out_size; (void)ws_size;
  const float* hidden = (const float*)d_in[0];   // [B,S,H]
  const int*   rels   = (const int*)d_in[1];     // [B,S,S]
  const float* basic  = (const float*)d_in[2];   // [4,H,H] == flat [3072, 768]
  const float* relw   = (const float*)d_in[3];   // [64,4]
  const float* selfw  = (const float*)d_in[4];   // [H,H] (out,in)
  float* out = (float*)d_out;                    // [B,S,H]

  float* ws   = (float*)d_ws;
  float* invf = ws;                              // B*S*32
  float* invr = invf + B_ * S_ * 32;             // B*S*32
  float* Wt   = invr + B_ * S_ * 32;             // H*H
  float* Wm   = Wt + H_ * H_;                    // B*4*S*S
  float* Z    = Wm + (long)B_ * 4 * S_ * S_;     // B*S*4*H   (total ~18.5 MB)

  count_kernel<<<B_ * S_, 256, 0, stream>>>(rels, invf, invr);
  wmat_kernel<<<(B_ * S_ * S_) / 256, 256, 0, stream>>>(rels, relw, invf, invr, Wm);
  transpose_kernel<<<dim3(H_ / 32, H_ / 32), dim3(32, 8), 0, stream>>>(selfw, Wt);
  gemm_stage1<<<dim3(H_ / 64, S_ / 64, B_ * 4), 128, 0, stream>>>(Wm, hidden, Z);
  gemm_stage2<<<dim3(H_ / 64, S_ / 64, B_), 128, 0, stream>>>(Z, basic, out);
  gemm_stage3<<<dim3(H_ / 64, S_ / 64, B_), 128, 0, stream>>>(hidden, Wt, out);
}